// WeightedChamferDistanceL2_33079838114187
// MI455X (gfx1250) — compile-verified
//
#include <hip/hip_runtime.h>
#include <hip/hip_bf16.h>

typedef __attribute__((ext_vector_type(2))) float v2f;
typedef __attribute__((ext_vector_type(8))) float v8f;

// D = A(16x4 f32) * B(4x16 f32) + C(16x16 f32), exact f32 math (K=3 padded with 0)
__device__ __forceinline__ v8f wmma_k4(v2f a, v2f b, v8f c) {
    // 8 args: (neg_a, A, neg_b, B, c_mod, C, reuse_a, reuse_b)
    return __builtin_amdgcn_wmma_f32_16x16x4_f32(
        false, a, false, b, (short)0, c, false, false);
}

// Load one 16-row A tile: fragment per ISA layout (lanes 0-15: K=0,1 ; lanes 16-31: K=2,3)
// and broadcast |x|^2 of rows (v + 8*hi) into sq[v] via cross-lane shuffle.
__device__ __forceinline__ void load_a_tile(const float* __restrict__ Xb, int base,
                                            int li, int hi, v2f& a, float sq[8]) {
    const int m = base + li;
    const float x0 = Xb[3 * m + 0];
    const float x1 = Xb[3 * m + 1];
    const float x2 = Xb[3 * m + 2];
    a[0] = hi ? x2 : x0;
    a[1] = hi ? 0.0f : x1;
    const float s = fmaf(x0, x0, fmaf(x1, x1, x2 * x2));
    // point m's norm lives in lane m (and m+16); row for VGPR v is (v + 8*hi)
#pragma unroll
    for (int v = 0; v < 8; ++v)
        sq[v] = __shfl(s, hi * 8 + v, 32);
}

// dist[b, i] = min_j || X[b,i,:] - Y[b,j,:] ||^2
// Grid: x = row tiles (256 rows per block: 8 waves * 32 rows), y = batch.
// nrows % 256 == 0, mcols % 16 == 0.
__global__ __launch_bounds__(256) void chamfer_min_kernel(
    const float* __restrict__ X, const float* __restrict__ Y,
    float* __restrict__ dist, int nrows, int mcols) {
    const int lane = threadIdx.x & 31;
    const int wave = threadIdx.x >> 5;
    const int li = lane & 15;
    const int hi = lane >> 4;
    const int b = blockIdx.y;

    const float* __restrict__ Xb = X + (size_t)b * nrows * 3;
    const float* __restrict__ Yb = Y + (size_t)b * mcols * 3;
    float* __restrict__ Db = dist + (size_t)b * nrows;

    const int row0 = (blockIdx.x * 8 + wave) * 32;  // two 16-row tiles per wave
    const int row1 = row0 + 16;

    v2f a0, a1;
    float sqx0[8], sqx1[8];
    load_a_tile(Xb, row0, li, hi, a0, sqx0);
    load_a_tile(Xb, row1, li, hi, a1, sqx1);

    v8f m0, m1;
#pragma unroll
    for (int v = 0; v < 8; ++v) { m0[v] = 3.4e38f; m1[v] = 3.4e38f; }

    const v8f czero = {};

#pragma unroll 2
    for (int t = 0; t < mcols; t += 16) {
        const int n = t + li;
        const float y0 = Yb[3 * n + 0];
        const float y1 = Yb[3 * n + 1];
        const float y2 = Yb[3 * n + 2];
        v2f bf;
        bf[0] = hi ? y2 : y0;
        bf[1] = hi ? 0.0f : y1;
        const float sqy = fmaf(y0, y0, fmaf(y1, y1, y2 * y2));  // lane's own D-column norm

        v8f c0 = wmma_k4(a0, bf, czero);
        v8f c1 = wmma_k4(a1, bf, czero);

#pragma unroll
        for (int v = 0; v < 8; ++v) {
            m0[v] = fminf(m0[v], fmaf(-2.0f, c0[v], sqx0[v] + sqy));
            m1[v] = fminf(m1[v], fmaf(-2.0f, c1[v], sqx1[v] + sqy));
        }
    }

    // min across the 16 lanes of each half (xor masks < 16 stay within the half)
#pragma unroll
    for (int v = 0; v < 8; ++v) {
#pragma unroll
        for (int mask = 1; mask < 16; mask <<= 1) {
            m0[v] = fminf(m0[v], __shfl_xor(m0[v], mask, 32));
            m1[v] = fminf(m1[v], __shfl_xor(m1[v], mask, 32));
        }
    }

    if (li == 0) {
#pragma unroll
        for (int v = 0; v < 8; ++v) {
            Db[row0 + 8 * hi + v] = m0[v];
            Db[row1 + 8 * hi + v] = m1[v];
        }
    }
}

// Single-block deterministic reduction:
// out = mean(dist_ic) + mean(dist_cp * dist_ci) / max(dist_cp)
__global__ __launch_bounds__(1024) void final_reduce_kernel(
    const float* __restrict__ dcp, const float* __restrict__ dic,
    const float* __restrict__ dci, float* __restrict__ out, int n) {
    __shared__ float s_ic[1024];
    __shared__ float s_pr[1024];
    __shared__ float s_mx[1024];

    const int tid = threadIdx.x;
    float sic = 0.0f, spr = 0.0f, mx = 0.0f;  // distances are >= 0
    for (int i = tid; i < n; i += 1024) {
        const float cp = dcp[i];
        sic += dic[i];
        spr = fmaf(cp, dci[i], spr);
        mx = fmaxf(mx, cp);
    }
    s_ic[tid] = sic;
    s_pr[tid] = spr;
    s_mx[tid] = mx;
    __syncthreads();

    for (int s = 512; s > 0; s >>= 1) {
        if (tid < s) {
            s_ic[tid] += s_ic[tid + s];
            s_pr[tid] += s_pr[tid + s];
            s_mx[tid] = fmaxf(s_mx[tid], s_mx[tid + s]);
        }
        __syncthreads();
    }

    if (tid == 0) {
        const float inv_n = 1.0f / (float)n;
        out[0] = s_ic[0] * inv_n + (s_pr[0] * inv_n) / s_mx[0];
    }
}

extern "C" void kernel_launch(void* const* d_in, const int* in_sizes, int n_in,
                              void* d_out, int out_size, void* d_ws, size_t ws_size,
                              hipStream_t stream) {
    (void)in_sizes; (void)n_in; (void)out_size; (void)ws_size;

    // setup_inputs order: partial [4,2048,3], infer [4,8192,3], complete [4,8192,3], fp32
    const float* partial  = (const float*)d_in[0];
    const float* infer    = (const float*)d_in[1];
    const float* complete = (const float*)d_in[2];
    float* out = (float*)d_out;

    constexpr int B = 4, NP = 2048, NC = 8192, NI = 8192;
    constexpr int NTOT = B * NC;  // 32768 (== B*NI)

    float* ws  = (float*)d_ws;
    float* dcp = ws;              // [B, NC]: per-complete min sq-dist to partial
    float* dic = ws + NTOT;       // [B, NI]: per-infer min sq-dist to complete
    float* dci = ws + 2 * NTOT;   // [B, NC]: per-complete min sq-dist to infer

    const dim3 blk(256);
    // rows per block = 256  ->  grid.x = nrows/256
    chamfer_min_kernel<<<dim3(NC / 256, B), blk, 0, stream>>>(complete, partial, dcp, NC, NP);
    chamfer_min_kernel<<<dim3(NI / 256, B), blk, 0, stream>>>(infer, complete, dic, NI, NC);
    chamfer_min_kernel<<<dim3(NC / 256, B), blk, 0, stream>>>(complete, infer, dci, NC, NI);

    final_reduce_kernel<<<1, 1024, 0, stream>>>(dcp, dic, dci, out, NTOT);
}